// AIF_2920577761873
// MI455X (gfx1250) — compile-verified
//
#include <hip/hip_runtime.h>
#include <hip/hip_bf16.h>
#include <math.h>

// Problem constants (match reference)
#define BB   16
#define TT   2048
#define VV   256
#define DENC 513
#define DD   512

typedef __attribute__((ext_vector_type(2))) float v2f;
typedef __attribute__((ext_vector_type(8))) float v8f;

#define WMMA_F32(a, b, c) \
  __builtin_amdgcn_wmma_f32_16x16x4_f32(false, (a), false, (b), (short)0, (c), false, false)

// ---------------------------------------------------------------------------
// Kernel 1: fire weights — alphas = sigmoid(xs[:,:,-1]) * enc_valid,
// inclusive cumsum over T, token_num = row sum. One block per batch.
// ---------------------------------------------------------------------------
__global__ void __launch_bounds__(256)
fire_kernel(const float* __restrict__ xs, const long long* __restrict__ xs_lens,
            float* __restrict__ alphas_out, float* __restrict__ csum_out,
            float* __restrict__ token_out) {
  __shared__ float sm[256];
  const int b   = blockIdx.x;
  const int tid = threadIdx.x;
  const long long xlen = xs_lens[b];
  const float* xrow = xs + (size_t)b * TT * DENC;

  float loc[8];
  float run = 0.0f;
  const int t0 = tid * 8;
#pragma unroll
  for (int i = 0; i < 8; ++i) {
    const int t = t0 + i;
    const float x = xrow[(size_t)t * DENC + (DENC - 1)];
    float a = 1.0f / (1.0f + __expf(-x));
    a = (t < (int)xlen) ? a : 0.0f;
    alphas_out[(size_t)b * TT + t] = a;
    run += a;
    loc[i] = run;  // inclusive prefix within thread
  }
  sm[tid] = run;
  __syncthreads();
  // Hillis-Steele inclusive scan over 256 per-thread totals
  for (int off = 1; off < 256; off <<= 1) {
    float v = (tid >= off) ? sm[tid - off] : 0.0f;
    __syncthreads();
    sm[tid] += v;
    __syncthreads();
  }
  const float excl = sm[tid] - run;
#pragma unroll
  for (int i = 0; i < 8; ++i)
    csum_out[(size_t)b * TT + t0 + i] = excl + loc[i];
  if (tid == 0) token_out[b] = sm[255];
}

// ---------------------------------------------------------------------------
// Kernel 2: xs_proj = xs[:,:,:512] @ W^T + b  -> P (B*T, 512), fp32 WMMA.
// Wave tile: 32(M) x 64(N): 2 A-fragments x 4 B-fragments -> 8 WMMAs/K-step.
// ---------------------------------------------------------------------------
__global__ void __launch_bounds__(256)
proj_kernel(const float* __restrict__ xs, const float* __restrict__ W,
            const float* __restrict__ bias, float* __restrict__ P) {
  const int lane  = threadIdx.x & 31;
  const int wid   = (blockIdx.x << 3) + (threadIdx.x >> 5);
  const int tileM = wid >> 3;         // (B*T)/32 = 1024 tiles
  const int tileN = wid & 7;          // 512/64   = 8 tiles
  const int m0 = tileM << 5;
  const int n0 = tileN << 6;
  const int row   = lane & 15;        // A: row M, B: col N
  const int khalf = (lane >> 4) << 1; // K sub-pair: 0 or 2

  const float* Arow0 = xs + (size_t)(m0 + row) * DENC;        // rows m0..m0+15
  const float* Arow1 = xs + (size_t)(m0 + 16 + row) * DENC;   // rows m0+16..m0+31
  const float* B0 = W + (size_t)(n0 +  0 + row) * (DENC - 1);
  const float* B1 = W + (size_t)(n0 + 16 + row) * (DENC - 1);
  const float* B2 = W + (size_t)(n0 + 32 + row) * (DENC - 1);
  const float* B3 = W + (size_t)(n0 + 48 + row) * (DENC - 1);

  v8f c00 = {}, c01 = {}, c02 = {}, c03 = {};
  v8f c10 = {}, c11 = {}, c12 = {}, c13 = {};
  for (int k = 0; k < DD; k += 4) {
    const int kk = k + khalf;
    v2f a0, a1;                       // xs row stride 513 is odd -> scalar loads
    a0.x = Arow0[kk]; a0.y = Arow0[kk + 1];
    a1.x = Arow1[kk]; a1.y = Arow1[kk + 1];
    v2f b0 = *(const v2f*)(B0 + kk);  // W stride 512 -> aligned b64 loads
    v2f b1 = *(const v2f*)(B1 + kk);
    v2f b2 = *(const v2f*)(B2 + kk);
    v2f b3 = *(const v2f*)(B3 + kk);
    c00 = WMMA_F32(a0, b0, c00);
    c01 = WMMA_F32(a0, b1, c01);
    c02 = WMMA_F32(a0, b2, c02);
    c03 = WMMA_F32(a0, b3, c03);
    c10 = WMMA_F32(a1, b0, c10);
    c11 = WMMA_F32(a1, b1, c11);
    c12 = WMMA_F32(a1, b2, c12);
    c13 = WMMA_F32(a1, b3, c13);
  }
  const int colBase = n0 + row;
  const int rowOff  = (lane >> 4) << 3;  // 0 or 8
  const float bb0 = bias[colBase +  0];
  const float bb1 = bias[colBase + 16];
  const float bb2 = bias[colBase + 32];
  const float bb3 = bias[colBase + 48];
#pragma unroll
  for (int r = 0; r < 8; ++r) {
    float* p0 = P + (size_t)(m0 + rowOff + r) * DD;
    float* p1 = P + (size_t)(m0 + 16 + rowOff + r) * DD;
    p0[colBase +  0] = c00[r] + bb0;
    p0[colBase + 16] = c01[r] + bb1;
    p0[colBase + 32] = c02[r] + bb2;
    p0[colBase + 48] = c03[r] + bb3;
    p1[colBase +  0] = c10[r] + bb0;
    p1[colBase + 16] = c11[r] + bb1;
    p1[colBase + 32] = c12[r] + bb2;
    p1[colBase + 48] = c13[r] + bb3;
  }
}

// ---------------------------------------------------------------------------
// Kernel 3: scores[b,v,t] = sum_d ys[b,v,d] * P[b,t,d]   (raw, unscaled)
// M=V=256, N=T=2048, K=D=512 per batch. Wave tile 32x64.
// ---------------------------------------------------------------------------
__global__ void __launch_bounds__(256)
scores_kernel(const float* __restrict__ ys, const float* __restrict__ P,
              float* __restrict__ S) {
  const int lane = threadIdx.x & 31;
  const int wid  = (blockIdx.x << 3) + (threadIdx.x >> 5);
  const int b = wid >> 8;     // 256 waves per batch
  const int r = wid & 255;
  const int tileM = r >> 5;   // 8 M-tiles (of 32 rows)
  const int tileN = r & 31;   // 32 N-tiles (of 64 cols)
  const int m0 = tileM << 5;
  const int n0 = tileN << 6;
  const int row   = lane & 15;
  const int khalf = (lane >> 4) << 1;

  const float* A0 = ys + (size_t)b * VV * DD + (size_t)(m0 + row) * DD;
  const float* A1 = A0 + (size_t)16 * DD;
  const float* Bm = P + (size_t)b * TT * DD;
  const float* B0 = Bm + (size_t)(n0 +  0 + row) * DD;
  const float* B1 = Bm + (size_t)(n0 + 16 + row) * DD;
  const float* B2 = Bm + (size_t)(n0 + 32 + row) * DD;
  const float* B3 = Bm + (size_t)(n0 + 48 + row) * DD;

  v8f c00 = {}, c01 = {}, c02 = {}, c03 = {};
  v8f c10 = {}, c11 = {}, c12 = {}, c13 = {};
  for (int k = 0; k < DD; k += 4) {
    const int kk = k + khalf;
    v2f a0 = *(const v2f*)(A0 + kk);
    v2f a1 = *(const v2f*)(A1 + kk);
    v2f b0 = *(const v2f*)(B0 + kk);
    v2f b1 = *(const v2f*)(B1 + kk);
    v2f b2 = *(const v2f*)(B2 + kk);
    v2f b3 = *(const v2f*)(B3 + kk);
    c00 = WMMA_F32(a0, b0, c00);
    c01 = WMMA_F32(a0, b1, c01);
    c02 = WMMA_F32(a0, b2, c02);
    c03 = WMMA_F32(a0, b3, c03);
    c10 = WMMA_F32(a1, b0, c10);
    c11 = WMMA_F32(a1, b1, c11);
    c12 = WMMA_F32(a1, b2, c12);
    c13 = WMMA_F32(a1, b3, c13);
  }
  float* Srow = S + (size_t)b * VV * TT;
  const int colBase = n0 + row;
  const int rowOff  = (lane >> 4) << 3;
#pragma unroll
  for (int rr = 0; rr < 8; ++rr) {
    float* o0 = Srow + (size_t)(m0 + rowOff + rr) * TT;
    float* o1 = Srow + (size_t)(m0 + 16 + rowOff + rr) * TT;
    o0[colBase +  0] = c00[rr];
    o0[colBase + 16] = c01[rr];
    o0[colBase + 32] = c02[rr];
    o0[colBase + 48] = c03[rr];
    o1[colBase +  0] = c10[rr];
    o1[colBase + 16] = c11[rr];
    o1[colBase + 32] = c12[rr];
    o1[colBase + 48] = c13[rr];
  }
}

// ---------------------------------------------------------------------------
// Kernel 4: masked, scaled softmax over T for each (b,v) row; in place on S.
// mask = (csum[b,t] <= v+1) && (t < xs_len[b]) && (v < ys_len[b]).
// ---------------------------------------------------------------------------
__global__ void __launch_bounds__(256)
softmax_kernel(const float* __restrict__ csum, const long long* __restrict__ xs_lens,
               const long long* __restrict__ ys_lens, float* __restrict__ S) {
  __shared__ float red[256];
  const int bv  = blockIdx.x;
  const int b   = bv >> 8;     // V = 256
  const int v   = bv & 255;
  const int tid = threadIdx.x;
  const float scale = 0.04419417382415922f;  // 1/sqrt(512)

  const long long xlen = xs_lens[b];
  const bool vvalid = v < (int)ys_lens[b];
  const float* cs = csum + (size_t)b * TT;
  float* row = S + (size_t)bv * TT;
  const float thr = (float)(v + 1);
  const float NEG = -__builtin_inff();

  float vals[8];
  float m = NEG;
#pragma unroll
  for (int i = 0; i < 8; ++i) {
    const int t = tid + (i << 8);
    const bool ok = vvalid && (t < (int)xlen) && (cs[t] <= thr);
    const float s = ok ? row[t] * scale : NEG;
    vals[i] = s;
    m = fmaxf(m, s);
  }
  red[tid] = m;
  __syncthreads();
  for (int off = 128; off > 0; off >>= 1) {
    if (tid < off) red[tid] = fmaxf(red[tid], red[tid + off]);
    __syncthreads();
  }
  m = red[0];
  __syncthreads();

  float lsum = 0.0f;
#pragma unroll
  for (int i = 0; i < 8; ++i) {
    const float e = (vals[i] == NEG) ? 0.0f : __expf(vals[i] - m);
    vals[i] = e;
    lsum += e;
  }
  red[tid] = lsum;
  __syncthreads();
  for (int off = 128; off > 0; off >>= 1) {
    if (tid < off) red[tid] += red[tid + off];
    __syncthreads();
  }
  const float total = red[0];
  const float inv = (m == NEG || total <= 0.0f) ? 0.0f : 1.0f / total;
#pragma unroll
  for (int i = 0; i < 8; ++i) row[tid + (i << 8)] = vals[i] * inv;
}

// ---------------------------------------------------------------------------
// Kernel 5: E[b,v,d] = sum_t attn[b,v,t] * P[b,t,d]
// M=V=256, N=D=512, K=T=2048 per batch. Wave tile 32x64.
// ---------------------------------------------------------------------------
__global__ void __launch_bounds__(256)
av_kernel(const float* __restrict__ S, const float* __restrict__ P,
          float* __restrict__ E) {
  const int lane = threadIdx.x & 31;
  const int wid  = (blockIdx.x << 3) + (threadIdx.x >> 5);
  const int b = wid >> 6;              // 64 waves per batch
  const int r = wid & 63;
  const int tileM = r >> 3;            // 8 M-tiles (of 32 rows)
  const int tileN = r & 7;             // 8 N-tiles (of 64 cols)
  const int m0 = tileM << 5;
  const int n0 = tileN << 6;
  const int row   = lane & 15;
  const int khalf = (lane >> 4) << 1;

  const float* A0 = S + (size_t)b * VV * TT + (size_t)(m0 + row) * TT;
  const float* A1 = A0 + (size_t)16 * TT;
  const float* Bm = P + (size_t)b * TT * DD;
  const int c0 = n0 + row;

  v8f c00 = {}, c01 = {}, c02 = {}, c03 = {};
  v8f c10 = {}, c11 = {}, c12 = {}, c13 = {};
  for (int k = 0; k < TT; k += 4) {
    const int kk = k + khalf;
    v2f a0 = *(const v2f*)(A0 + kk);   // attn rows, stride 2048 -> aligned
    v2f a1 = *(const v2f*)(A1 + kk);
    const float* Bk0 = Bm + (size_t)kk * DD;   // row kk of P
    const float* Bk1 = Bk0 + DD;               // row kk+1
    v2f b0, b1, b2, b3;
    b0.x = Bk0[c0 +  0]; b0.y = Bk1[c0 +  0];
    b1.x = Bk0[c0 + 16]; b1.y = Bk1[c0 + 16];
    b2.x = Bk0[c0 + 32]; b2.y = Bk1[c0 + 32];
    b3.x = Bk0[c0 + 48]; b3.y = Bk1[c0 + 48];
    c00 = WMMA_F32(a0, b0, c00);
    c01 = WMMA_F32(a0, b1, c01);
    c02 = WMMA_F32(a0, b2, c02);
    c03 = WMMA_F32(a0, b3, c03);
    c10 = WMMA_F32(a1, b0, c10);
    c11 = WMMA_F32(a1, b1, c11);
    c12 = WMMA_F32(a1, b2, c12);
    c13 = WMMA_F32(a1, b3, c13);
  }
  const int rowOff = (lane >> 4) << 3;
#pragma unroll
  for (int rr = 0; rr < 8; ++rr) {
    float* o0 = E + (size_t)b * VV * DD + (size_t)(m0 + rowOff + rr) * DD;
    float* o1 = E + (size_t)b * VV * DD + (size_t)(m0 + 16 + rowOff + rr) * DD;
    o0[c0 +  0] = c00[rr];
    o0[c0 + 16] = c01[rr];
    o0[c0 + 32] = c02[rr];
    o0[c0 + 48] = c03[rr];
    o1[c0 +  0] = c10[rr];
    o1[c0 + 16] = c11[rr];
    o1[c0 + 32] = c12[rr];
    o1[c0 + 48] = c13[rr];
  }
}

// ---------------------------------------------------------------------------
extern "C" void kernel_launch(void* const* d_in, const int* in_sizes, int n_in,
                              void* d_out, int out_size, void* d_ws, size_t ws_size,
                              hipStream_t stream) {
  const float*     xs      = (const float*)d_in[0];      // (B,T,DENC)
  const float*     ys      = (const float*)d_in[1];      // (B,V,D)
  const float*     W       = (const float*)d_in[2];      // (D, DENC-1)
  const float*     bias    = (const float*)d_in[3];      // (D,)
  const long long* xs_lens = (const long long*)d_in[4];  // (B,) int64
  const long long* ys_lens = (const long long*)d_in[5];  // (B,) int64

  // Output layout: [embeds B*V*D][token_num B][alphas B*T][csum B*T]
  float* emb    = (float*)d_out;
  float* token  = emb + (size_t)BB * VV * DD;
  float* alphas = token + BB;
  float* csum   = alphas + (size_t)BB * TT;

  // Workspace: P (B*T*D fp32, 64 MiB) then S (B*V*T fp32, 32 MiB)
  float* P = (float*)d_ws;
  float* S = P + (size_t)BB * TT * DD;

  // 1) fire weights + cumsum (one block per batch)
  fire_kernel<<<BB, 256, 0, stream>>>(xs, xs_lens, alphas, csum, token);

  // 2) projection GEMM: ((B*T)/32)*(512/64) = 8192 waves -> 1024 blocks
  proj_kernel<<<1024, 256, 0, stream>>>(xs, W, bias, P);

  // 3) scores GEMM: 16 * 256 = 4096 waves -> 512 blocks
  scores_kernel<<<512, 256, 0, stream>>>(ys, P, S);

  // 4) masked softmax: one block per (b,v) row
  softmax_kernel<<<BB * VV, 256, 0, stream>>>(csum, xs_lens, ys_lens, S);

  // 5) attn @ P: 16 * 64 = 1024 waves -> 128 blocks
  av_kernel<<<128, 256, 0, stream>>>(S, P, emb);
}